// GCNReLU_44796508897660
// MI455X (gfx1250) — compile-verified
//
#include <hip/hip_runtime.h>

// GCN 2-layer forward for MI455X (gfx1250, wave32).
// GEMMs use V_WMMA_F32_16X16X4_F32 (full fp32 precision, 16x16 tile per wave).
// Edge aggregation uses float4 gathers + global f32 atomics (L2-resident accumulator).
// BatchNorm: fused combine + per-column partial stats -> 128-entry atomics -> scale/shift.

typedef __attribute__((ext_vector_type(2))) float v2f;
typedef __attribute__((ext_vector_type(8))) float v8f;

#define D 128

// ---------------- zero fill ----------------
__global__ void zero_f4_kernel(float4* __restrict__ p, long n4) {
    long i = blockIdx.x * (long)blockDim.x + threadIdx.x;
    if (i < n4) p[i] = make_float4(0.f, 0.f, 0.f, 0.f);
}

// ---------------- dual GEMM: M = X@W ; R = relu(X@Wr + br) ----------------
// block (32,4): 4 waves, each wave computes one 16x16 tile of both products.
// grid (2, N/16): x covers 8 col-tiles (D=128), y covers row tiles.
// Requires N % 16 == 0 (true: N = 100000).
__global__ __launch_bounds__(128)
void gemm_dual_kernel(const float* __restrict__ X,
                      const float* __restrict__ W,
                      const float* __restrict__ Wr,
                      const float* __restrict__ brv,
                      float* __restrict__ M,
                      float* __restrict__ R,
                      int N)
{
    const int lane = threadIdx.x;        // 0..31
    const int l    = lane & 15;
    const int half = lane >> 4;          // 0 or 1
    const int c0   = (blockIdx.x * 4 + threadIdx.y) * 16;
    const int r0   = blockIdx.y * 16;

    const float* xrow = X + (long)(r0 + l) * D;  // A-frag row for this lane

    v8f acc  = {};
    v8f accr = {};

    #pragma unroll
    for (int k0 = 0; k0 < D; k0 += 4) {
        const int ka = k0 + half * 2;
        // A fragment (16x4 f32): lanes 0-15 hold K=k0,k0+1; lanes 16-31 hold K=k0+2,k0+3
        v2f a;
        a.x = xrow[ka];
        a.y = xrow[ka + 1];
        // B fragment (4x16 f32), mirrored layout: lane -> col, vgpr -> K within half
        v2f b;
        b.x = W[(long)ka * D + c0 + l];
        b.y = W[(long)(ka + 1) * D + c0 + l];
        v2f b2;
        b2.x = Wr[(long)ka * D + c0 + l];
        b2.y = Wr[(long)(ka + 1) * D + c0 + l];

        acc  = __builtin_amdgcn_wmma_f32_16x16x4_f32(false, a, false, b,  (short)0, acc,  false, false);
        accr = __builtin_amdgcn_wmma_f32_16x16x4_f32(false, a, false, b2, (short)0, accr, false, false);
    }

    // C/D layout: vgpr i -> row (r0 + half*8 + i), col (c0 + l)
    const int col  = c0 + l;
    const int rowb = r0 + half * 8;
    const float bb = brv[col];
    #pragma unroll
    for (int i = 0; i < 8; ++i) {
        long idx = (long)(rowb + i) * D + col;
        M[idx] = acc[i];
        float v = accr[i] + bb;
        R[idx] = v > 0.f ? v : 0.f;
    }
}

// ---------------- edge scatter: AGG[dst] += M[src] ----------------
// one thread per (edge, 4-float chunk): 32 chunks per edge row of 128 floats.
__global__ void scatter_add_kernel(const float* __restrict__ M,
                                   const int* __restrict__ src,
                                   const int* __restrict__ dst,
                                   float* __restrict__ AGG,
                                   long E)
{
    long t = blockIdx.x * (long)blockDim.x + threadIdx.x;
    long total = E << 5;
    if (t >= total) return;
    long e = t >> 5;
    int  q = (int)(t & 31) * 4;
    int  s = src[e];
    int  d = dst[e];
    const float4 v = *(const float4*)(M + (long)s * D + q);
    float* o = AGG + (long)d * D + q;
    atomicAdd(o + 0, v.x);
    atomicAdd(o + 1, v.y);
    atomicAdd(o + 2, v.z);
    atomicAdd(o + 3, v.w);
}

// ---------------- combine + column stats ----------------
// OUT = relu(AGG + bias) + R ; accumulate per-column sum / sumsq.
// 128 threads = one column each; 128 rows per block (coalesced row reads).
__global__ __launch_bounds__(128)
void combine_stats_kernel(const float* __restrict__ AGG,
                          const float* __restrict__ R,
                          const float* __restrict__ bias,
                          float* __restrict__ OUT,
                          float* __restrict__ csum,
                          float* __restrict__ csq,
                          int N)
{
    const int col  = threadIdx.x;
    const int row0 = blockIdx.x * 128;
    const float b  = bias[col];
    float s = 0.f, s2 = 0.f;
    #pragma unroll 4
    for (int i = 0; i < 128; ++i) {
        int row = row0 + i;
        if (row >= N) break;
        long idx = (long)row * D + col;
        float v = AGG[idx] + b;
        v = v > 0.f ? v : 0.f;
        v += R[idx];
        OUT[idx] = v;
        s  += v;
        s2 += v * v;
    }
    atomicAdd(&csum[col], s);
    atomicAdd(&csq[col],  s2);
}

// ---------------- BN prep: scale/shift from population stats ----------------
__global__ void bn_prep_kernel(const float* __restrict__ csum,
                               const float* __restrict__ csq,
                               const float* __restrict__ g,
                               const float* __restrict__ be,
                               float* __restrict__ scale,
                               float* __restrict__ shift,
                               int N)
{
    int c = threadIdx.x;
    float inv = 1.f / (float)N;
    float mu  = csum[c] * inv;
    float var = csq[c] * inv - mu * mu;
    float sc  = g[c] * rsqrtf(var + 1e-5f);
    scale[c] = sc;
    shift[c] = be[c] - mu * sc;
}

// ---------------- BN apply: Y = X*scale[col] + shift[col] ----------------
__global__ void bn_apply_kernel(const float* __restrict__ X,
                                const float* __restrict__ scale,
                                const float* __restrict__ shift,
                                float* __restrict__ Y,
                                long n4)
{
    long t = blockIdx.x * (long)blockDim.x + threadIdx.x;
    if (t >= n4) return;
    int q = (int)(t & 31) * 4;   // column of first element of this float4
    float4 v = ((const float4*)X)[t];
    v.x = v.x * scale[q + 0] + shift[q + 0];
    v.y = v.y * scale[q + 1] + shift[q + 1];
    v.z = v.z * scale[q + 2] + shift[q + 2];
    v.w = v.w * scale[q + 3] + shift[q + 3];
    ((float4*)Y)[t] = v;
}

extern "C" void kernel_launch(void* const* d_in, const int* in_sizes, int n_in,
                              void* d_out, int out_size, void* d_ws, size_t ws_size,
                              hipStream_t stream) {
    const float* h   = (const float*)d_in[0];
    const int*   src = (const int*)  d_in[1];
    const int*   dst = (const int*)  d_in[2];
    const float* W0  = (const float*)d_in[3];
    const float* b0  = (const float*)d_in[4];
    const float* Wr0 = (const float*)d_in[5];
    const float* br0 = (const float*)d_in[6];
    const float* g0  = (const float*)d_in[7];
    const float* be0 = (const float*)d_in[8];
    const float* W1  = (const float*)d_in[9];
    const float* b1  = (const float*)d_in[10];
    const float* Wr1 = (const float*)d_in[11];
    const float* br1 = (const float*)d_in[12];
    const float* g1  = (const float*)d_in[13];
    const float* be1 = (const float*)d_in[14];

    const int  N = in_sizes[0] / D;        // 100000
    const long E = (long)in_sizes[1];      // 1600000

    // workspace: three N*D buffers + 512 floats of stats
    float* A    = (float*)d_ws;
    float* B    = A + (long)N * D;
    float* C    = B + (long)N * D;
    float* csum  = C + (long)N * D;
    float* csq   = csum + D;
    float* scale = csq + D;
    float* shift = scale + D;

    float* out = (float*)d_out;

    const dim3 gblk(32, 4);
    const dim3 ggrd(2, (N + 15) / 16);
    const long n4 = (long)N * 32;                      // float4 count of an N*D buffer
    const long zg = (n4 + 255) / 256;
    const long st = E << 5;                            // scatter threads
    const long sg = (st + 255) / 256;
    const int  cg = (N + 127) / 128;

    // ---- Layer 1: X=h, M->A, R->B, AGG->C, OUT->A, Y1->B ----
    zero_f4_kernel<<<zg, 256, 0, stream>>>((float4*)C, n4);
    zero_f4_kernel<<<1, 64, 0, stream>>>((float4*)csum, 64);   // csum + csq
    gemm_dual_kernel<<<ggrd, gblk, 0, stream>>>(h, W0, Wr0, br0, A, B, N);
    scatter_add_kernel<<<sg, 256, 0, stream>>>(A, src, dst, C, E);
    combine_stats_kernel<<<cg, 128, 0, stream>>>(C, B, b0, A, csum, csq, N);
    bn_prep_kernel<<<1, 128, 0, stream>>>(csum, csq, g0, be0, scale, shift, N);
    bn_apply_kernel<<<zg, 256, 0, stream>>>(A, scale, shift, B, n4);

    // ---- Layer 2: X=B, M->A, R->C, AGG->B (after gemm), OUT->A, Y->d_out ----
    gemm_dual_kernel<<<ggrd, gblk, 0, stream>>>(B, W1, Wr1, br1, A, C, N);
    zero_f4_kernel<<<zg, 256, 0, stream>>>((float4*)B, n4);    // B free after gemm read
    zero_f4_kernel<<<1, 64, 0, stream>>>((float4*)csum, 64);
    scatter_add_kernel<<<sg, 256, 0, stream>>>(A, src, dst, B, E);
    combine_stats_kernel<<<cg, 128, 0, stream>>>(B, C, b1, A, csum, csq, N);
    bn_prep_kernel<<<1, 128, 0, stream>>>(csum, csq, g1, be1, scale, shift, N);
    bn_apply_kernel<<<zg, 256, 0, stream>>>(A, scale, shift, out, n4);
}